// MultiHeadGATLayer_66262755443073
// MI455X (gfx1250) — compile-verified
//
#include <hip/hip_runtime.h>
#include <hip/hip_bf16.h>
#include <stdint.h>

// Problem constants (fixed by the reference): IN = H*D = 256, H = 4, D = 64
#define HD        256
#define NHEAD     4
#define DDIM      64
#define K_IN      256
#define NEG_SLOPE 0.2f

typedef __attribute__((ext_vector_type(2))) float v2f;
typedef __attribute__((ext_vector_type(8))) float v8f;

// ---------------------------------------------------------------------------
// 1) h = x @ W   via V_WMMA_F32_16X16X4_F32, one wave per 16x16 tile of h.
// A (16x4 fp32): lanes 0-15 hold K=(k,k+1), lanes 16-31 hold K=(k+2,k+3),
//                row M = lane&15, K pair across 2 VGPRs.
// B (4x16 fp32): lanes 0-15 hold rows K=(k,k+1), lanes 16-31 rows K=(k+2,k+3),
//                col N = lane&15.
// D (16x16 fp32, 8 VGPRs): VGPR r -> row r (lanes 0-15) / row 8+r (lanes 16-31),
//                col = lane&15.
// ---------------------------------------------------------------------------
__global__ void gat_gemm_wmma(const float* __restrict__ x,
                              const float* __restrict__ W,
                              float* __restrict__ h,
                              int N, int nTilesRow) {
  const int wavesPerBlock = blockDim.x >> 5;
  const int wave = blockIdx.x * wavesPerBlock + (threadIdx.x >> 5);
  const int nTiles = nTilesRow * (HD / 16);
  if (wave >= nTiles) return;

  const int rowTile = wave / (HD / 16);
  const int colTile = wave % (HD / 16);
  const int lane    = threadIdx.x & 31;
  const int halfSel = lane >> 4;     // 0: lanes 0-15, 1: lanes 16-31
  const int mr      = lane & 15;

  int row = rowTile * 16 + mr;
  if (row >= N) row = N - 1;         // clamp loads for ragged tail (store is guarded)
  const int col = colTile * 16 + mr;

  const float* __restrict__ xrow = x + (size_t)row * K_IN;

  v8f acc = {};
  #pragma unroll 8
  for (int k = 0; k < K_IN; k += 4) {
    const int ka = k + 2 * halfSel;
    v2f a = *(const v2f*)(xrow + ka);          // b64 load of the lane's K pair
    v2f b;
    b.x = W[(size_t)(ka + 0) * HD + col];      // coalesced across lanes 0-15
    b.y = W[(size_t)(ka + 1) * HD + col];
    acc = __builtin_amdgcn_wmma_f32_16x16x4_f32(
        /*neg_a=*/false, a, /*neg_b=*/false, b,
        /*c_mod=*/(short)0, acc, /*reuse_a=*/false, /*reuse_b=*/false);
  }

  // Epilogue: wave-uniform interior/tail split keeps the common path branch-free
  // (avoids 8 exec-mask save/cmpx/restore sequences per tile).
  const int baseRow = rowTile * 16 + 8 * halfSel;
  const int ocol    = colTile * 16 + (lane & 15);
  float* __restrict__ hp = h + (size_t)baseRow * HD + ocol;
  if (rowTile * 16 + 16 <= N) {                // fully interior tile (always, here)
    #pragma unroll
    for (int r = 0; r < 8; ++r) {
      hp[(size_t)r * HD] = acc[r];
    }
  } else {                                     // ragged tail tile
    #pragma unroll
    for (int r = 0; r < 8; ++r) {
      if (baseRow + r < N) hp[(size_t)r * HD] = acc[r];
    }
  }
}

// ---------------------------------------------------------------------------
// 2) Per-node, per-head attention logits: a_s[n,h], a_d[n,h]
// ---------------------------------------------------------------------------
__global__ void gat_logits(const float* __restrict__ h,
                           const float* __restrict__ att_src,
                           const float* __restrict__ att_dst,
                           float* __restrict__ a_s,
                           float* __restrict__ a_d, int N) {
  const int idx = blockIdx.x * blockDim.x + threadIdx.x;   // n*NHEAD + head
  if (idx >= N * NHEAD) return;
  const int n  = idx >> 2;
  const int hd = idx & 3;
  const float* __restrict__ hp = h + (size_t)n * HD + hd * DDIM;
  const float* __restrict__ as = att_src + hd * DDIM;
  const float* __restrict__ ad = att_dst + hd * DDIM;
  float ss = 0.f, sd = 0.f;
  #pragma unroll 8
  for (int d = 0; d < DDIM; ++d) {
    const float v = hp[d];
    ss = fmaf(v, as[d], ss);
    sd = fmaf(v, ad[d], sd);
  }
  a_s[idx] = ss;
  a_d[idx] = sd;
}

// ---------------------------------------------------------------------------
// 3) Init: out = bias (broadcast), m = 0 (ordered-uint -inf sentinel), s = 0
// ---------------------------------------------------------------------------
__global__ void gat_init(float* __restrict__ out, const float* __restrict__ bias,
                         unsigned* __restrict__ m, float* __restrict__ s, int N) {
  const int idx = blockIdx.x * blockDim.x + threadIdx.x;
  if (idx < N * HD) out[idx] = bias[idx & (HD - 1)];
  if (idx < N * NHEAD) { m[idx] = 0u; s[idx] = 0.f; }
}

// Monotone float <-> uint mapping so atomicMax(u32) implements float max.
__device__ __forceinline__ unsigned f2ord(float f) {
  const unsigned u = __float_as_uint(f);
  return (u & 0x80000000u) ? ~u : (u | 0x80000000u);
}
__device__ __forceinline__ float ord2f(unsigned e) {
  const unsigned b = (e & 0x80000000u) ? (e ^ 0x80000000u) : ~e;
  return __uint_as_float(b);
}

__device__ __forceinline__ void edge_endpoints(const long long* __restrict__ ei,
                                               long long e, long long E,
                                               int& src, int& dst) {
  if (e < E) { src = (int)ei[e]; dst = (int)ei[E + e]; }
  else       { src = dst = (int)(e - E); }            // self-loops appended
}

// ---------------------------------------------------------------------------
// 4) Pass 1: e = leaky_relu(a_s[src]+a_d[dst]); segment max over dst
// ---------------------------------------------------------------------------
__global__ void gat_edge_max(const long long* __restrict__ ei,
                             const float* __restrict__ a_s,
                             const float* __restrict__ a_d,
                             unsigned* __restrict__ m,
                             long long E, int N) {
  const long long idx = (long long)blockIdx.x * blockDim.x + threadIdx.x;
  const long long EP = E + N;
  if (idx >= EP * NHEAD) return;
  const long long e = idx >> 2;
  const int hd = (int)(idx & 3);
  int src, dst;
  edge_endpoints(ei, e, E, src, dst);
  float v = a_s[src * NHEAD + hd] + a_d[dst * NHEAD + hd];
  v = (v > 0.f) ? v : NEG_SLOPE * v;
  atomicMax(&m[dst * NHEAD + hd], f2ord(v));
}

// ---------------------------------------------------------------------------
// 5) Pass 2: w = exp(e - m[dst]); segment sum over dst; stash w
// ---------------------------------------------------------------------------
__global__ void gat_edge_expsum(const long long* __restrict__ ei,
                                const float* __restrict__ a_s,
                                const float* __restrict__ a_d,
                                const unsigned* __restrict__ m,
                                float* __restrict__ s,
                                float* __restrict__ w,
                                long long E, int N) {
  const long long idx = (long long)blockIdx.x * blockDim.x + threadIdx.x;
  const long long EP = E + N;
  if (idx >= EP * NHEAD) return;
  const long long e = idx >> 2;
  const int hd = (int)(idx & 3);
  int src, dst;
  edge_endpoints(ei, e, E, src, dst);
  float v = a_s[src * NHEAD + hd] + a_d[dst * NHEAD + hd];
  v = (v > 0.f) ? v : NEG_SLOPE * v;
  const float wv = __expf(v - ord2f(m[dst * NHEAD + hd]));
  w[idx] = wv;
  atomicAdd(&s[dst * NHEAD + hd], wv);
}

// ---------------------------------------------------------------------------
// 6) Pass 3: out[dst] += (w/s[dst]) * h[src]   (one 256-thread block per edge;
//    thread t owns feature t, head = t>>6). h is L2-resident (51MB << 192MB).
// ---------------------------------------------------------------------------
__global__ void gat_aggregate(const long long* __restrict__ ei,
                              const float* __restrict__ h,
                              const float* __restrict__ w,
                              const float* __restrict__ s,
                              float* __restrict__ out,
                              long long E, int N) {
  const long long e = blockIdx.x;
  const int t  = threadIdx.x;
  const int hd = t >> 6;
  int src, dst;
  edge_endpoints(ei, e, E, src, dst);
  const float alpha = w[e * NHEAD + hd] / s[dst * NHEAD + hd];
  const float val = alpha * h[(size_t)src * HD + t];
  atomicAdd(&out[(size_t)dst * HD + t], val);
}

// ---------------------------------------------------------------------------
extern "C" void kernel_launch(void* const* d_in, const int* in_sizes, int n_in,
                              void* d_out, int out_size, void* d_ws, size_t ws_size,
                              hipStream_t stream) {
  const float*     x       = (const float*)d_in[0];
  const long long* ei      = (const long long*)d_in[1];   // int64 [2,E]
  const float*     W       = (const float*)d_in[2];
  const float*     att_src = (const float*)d_in[3];
  const float*     att_dst = (const float*)d_in[4];
  const float*     bias    = (const float*)d_in[5];
  float*           out     = (float*)d_out;

  const int       N  = in_sizes[0] / K_IN;     // 50000
  const long long E  = in_sizes[1] / 2;        // 800000
  const long long EP = E + N;

  // Workspace carve-up
  float*    h   = (float*)d_ws;                          // N*HD
  float*    a_s = h + (size_t)N * HD;                    // N*NHEAD
  float*    a_d = a_s + (size_t)N * NHEAD;               // N*NHEAD
  unsigned* m   = (unsigned*)(a_d + (size_t)N * NHEAD);  // N*NHEAD
  float*    s   = (float*)(m + (size_t)N * NHEAD);       // N*NHEAD
  float*    w   = s + (size_t)N * NHEAD;                 // EP*NHEAD

  // 1) GEMM via WMMA: 8 waves (256 threads) per block, one 16x16 tile per wave
  const int nTilesRow = (N + 15) / 16;
  const int nTiles    = nTilesRow * (HD / 16);
  gat_gemm_wmma<<<(nTiles + 7) / 8, 256, 0, stream>>>(x, W, h, N, nTilesRow);

  // 2) Attention logits
  gat_logits<<<(N * NHEAD + 255) / 256, 256, 0, stream>>>(h, att_src, att_dst,
                                                          a_s, a_d, N);

  // 3) Init out/m/s
  gat_init<<<((size_t)N * HD + 255) / 256, 256, 0, stream>>>(out, bias, m, s, N);

  // 4-5) Edge softmax (max, then exp+sum)
  const long long nEdgeWork = EP * NHEAD;
  const int edgeBlocks = (int)((nEdgeWork + 255) / 256);
  gat_edge_max<<<edgeBlocks, 256, 0, stream>>>(ei, a_s, a_d, m, E, N);
  gat_edge_expsum<<<edgeBlocks, 256, 0, stream>>>(ei, a_s, a_d, m, s, w, E, N);

  // 6) Weighted scatter-aggregate
  gat_aggregate<<<(unsigned)EP, 256, 0, stream>>>(ei, h, w, s, out, E, N);
}